// Discriminator_cluster_38517266711073
// MI455X (gfx1250) — compile-verified
//
#include <hip/hip_runtime.h>
#include <hip/hip_bf16.h>

typedef __attribute__((ext_vector_type(2))) float v2f;
typedef __attribute__((ext_vector_type(8))) float v8f;

#define EPS 1e-8f
#define WAVES_PER_BLOCK 4

// ---------------------------------------------------------------------------
// Kernel 1: inv_cnorm = 1 / sqrt(sum(c*c) + nh*eps)   (single block)
// ---------------------------------------------------------------------------
__global__ void cnorm_kernel(const float* __restrict__ c, float* __restrict__ inv_out, int nh) {
    __shared__ float red[256];
    float s = 0.0f;
    for (int i = threadIdx.x; i < nh; i += blockDim.x) {
        float v = c[i];
        s = __builtin_fmaf(v, v, s);
    }
    red[threadIdx.x] = s;
    __syncthreads();
    for (int w = 128; w > 0; w >>= 1) {
        if ((int)threadIdx.x < w) red[threadIdx.x] += red[threadIdx.x + w];
        __syncthreads();
    }
    if (threadIdx.x == 0) inv_out[0] = 1.0f / sqrtf(red[0] + (float)nh * EPS);
}

// ---------------------------------------------------------------------------
// Kernel 2: per-row cosine vs c, dot products via V_WMMA_F32_16X16X4_F32.
//   One wave handles 16 rows.  Per K-chunk of 4:
//     A (16x4 f32): lane L -> row (L%16), elems k + (L/16)*2 + {0,1}
//     B (4x16 f32): same offsets loaded from c -> every column of B == c chunk
//   => every column of D accumulates dot(row_m, c).  Sum-of-squares is done
//   with per-lane FMAs (row m is split across lanes m and m+16).
// ---------------------------------------------------------------------------
__global__ void cosine_wmma_kernel(const float* __restrict__ c,
                                   const float* __restrict__ h_pl,
                                   const float* __restrict__ h_mi,
                                   const float* __restrict__ inv_cnorm_p,
                                   float* __restrict__ out,
                                   int n_nb, int nh, int ntiles) {
    __shared__ float s_dot[WAVES_PER_BLOCK][16];
    __shared__ float s_sq[WAVES_PER_BLOCK][32];

    const int lane = threadIdx.x & 31;
    const int wave = threadIdx.x >> 5;
    const int m    = lane & 15;   // row within tile
    const int half = lane >> 4;   // which K-half of the chunk this lane loads

    const int  tile   = blockIdx.x * WAVES_PER_BLOCK + wave;
    const bool active = tile < ntiles;            // uniform per wave
    const long g      = (long)(active ? tile : 0) * 16 + m;  // global row
    const float* row  = (g < (long)n_nb) ? (h_pl + g * (long)nh)
                                         : (h_mi + (g - n_nb) * (long)nh);

    v8f   acc = {};   // D[m,n] = dot(row_m, c) in every column n
    float sq  = 0.0f; // per-lane partial sum of squares

#pragma unroll 4
    for (int k = 0; k < nh; k += 4) {
        const int ko = k + half * 2;
        v2f a  = __builtin_nontemporal_load((const v2f*)(row + ko)); // streaming
        v2f bc = *(const v2f*)(c + ko);                              // cache-resident
        // D = A x B + C  (f32, 16x16x4)
        acc = __builtin_amdgcn_wmma_f32_16x16x4_f32(
            /*neg_a=*/false, a, /*neg_b=*/false, bc,
            /*c_mod=*/(short)0, acc, /*reuse_a=*/false, /*reuse_b=*/false);
        sq = __builtin_fmaf(a.x, a.x, sq);
        sq = __builtin_fmaf(a.y, a.y, sq);
    }

    // Stage cross-lane pieces through per-wave LDS.
    // D layout: VGPR r, lanes 0-15 -> M=r (N=lane), lanes 16-31 -> M=r+8.
    // Column 0 lives in lanes 0 (rows 0..7) and 16 (rows 8..15).
    s_sq[wave][lane] = sq;
    if (m == 0) {
#pragma unroll
        for (int r = 0; r < 8; ++r) s_dot[wave][half * 8 + r] = acc[r];
    }
    __syncthreads();

    if (active && lane < 16) {
        float ssum   = s_sq[wave][lane] + s_sq[wave][lane + 16]; // full row sumsq
        float xnorm  = sqrtf(ssum + (float)nh * EPS);
        float result = (s_dot[wave][lane] / xnorm) * inv_cnorm_p[0];
        __builtin_nontemporal_store(result, out + g);
    }
}

// ---------------------------------------------------------------------------
extern "C" void kernel_launch(void* const* d_in, const int* in_sizes, int n_in,
                              void* d_out, int out_size, void* d_ws, size_t ws_size,
                              hipStream_t stream) {
    const float* c    = (const float*)d_in[0];
    // d_in[1] (h_0) is unused by the reference output.
    const float* h_pl = (const float*)d_in[2];
    const float* h_mi = (const float*)d_in[3];
    float* out = (float*)d_out;
    float* ws  = (float*)d_ws;   // ws[0] = 1/||c||

    const int nh     = in_sizes[0];        // 512
    const int n_nb   = in_sizes[2] / nh;   // 100000
    const int total  = 2 * n_nb;           // == out_size
    const int ntiles = (total + 15) / 16;  // 16 rows per wave
    const int blocks = (ntiles + WAVES_PER_BLOCK - 1) / WAVES_PER_BLOCK;

    cnorm_kernel<<<1, 256, 0, stream>>>(c, ws, nh);
    cosine_wmma_kernel<<<blocks, WAVES_PER_BLOCK * 32, 0, stream>>>(
        c, h_pl, h_mi, ws, out, n_nb, nh, ntiles);
}